// HistogramMatcher_22703197126822
// MI455X (gfx1250) — compile-verified
//
#include <hip/hip_runtime.h>
#include <stdint.h>

// Histogram matching, 4096x4096x3 f32. Memory-bound (~800MB -> ~35us @23.3TB/s).
// CDNA5 paths used: ds_add_u32 LDS-privatized histograms, global_prefetch_b8,
// global_load_async_to_lds_b128 + s_wait_asynccnt for LDS table staging.

#define NBINS 256

static constexpr int HIST_WORDS = 6 * NBINS;           // u32: [src c0..2 | tgt c0..2]
static constexpr int DXS_F  = 3 * NBINS;               // cdf_src knots per channel
static constexpr int TAB_F  = 3 * NBINS * 4;           // {x0,denom,dy0,ddy} per bin
static constexpr int EDGE_F = 8;                       // pm[0],pm[255] per channel (+pad)
static constexpr int TBL_F  = DXS_F + TAB_F + EDGE_F;  // 3848 floats = 15392 B (962 x b128)

__device__ __forceinline__ int binof(float v) {
  int b = (int)((v + 1.0f) * 128.0f);          // trunc-toward-zero == jnp astype(int32)
  return b < 0 ? 0 : (b > 255 ? 255 : b);      // == jnp.clip(.,0,255)
}

// searchsorted(a, x, side='left') over 256 sorted elements: count of a[i] < x.
__device__ __forceinline__ unsigned lb256(const float* a, float x) {
  unsigned p = 0;
#pragma unroll
  for (unsigned s = 128; s; s >>= 1) {
    unsigned q = p + s;
    if (a[q - 1] < x) p = q;
  }
  return p;                                    // in [0,256]
}

__global__ __launch_bounds__(256) void hm_zero(unsigned* __restrict__ hist) {
  int i = blockIdx.x * 256 + threadIdx.x;
  if (i < HIST_WORDS) hist[i] = 0u;
}

__global__ __launch_bounds__(256) void hm_hist(const float* __restrict__ src,
                                               const float* __restrict__ tgt,
                                               unsigned* __restrict__ hist,
                                               int n) {
  __shared__ unsigned h[HIST_WORDS];
  for (int i = threadIdx.x; i < HIST_WORDS; i += blockDim.x) h[i] = 0u;
  __syncthreads();
  const int stride = gridDim.x * blockDim.x;
  const int tid = blockIdx.x * blockDim.x + threadIdx.x;
  const int nf4 = n >> 2;
  const float4* s4 = (const float4*)src;
  const float4* t4 = (const float4*)tgt;
  for (int i = tid; i < nf4; i += stride) {
    __builtin_prefetch(&s4[i + stride], 0, 0);   // global_prefetch_b8 (speculative)
    __builtin_prefetch(&t4[i + stride], 0, 0);
    float4 a = s4[i];
    float4 b = t4[i];
    int c0 = i % 3;                              // channel of element 4i (4 == 1 mod 3)
    int c1 = c0 + 1; if (c1 == 3) c1 = 0;
    int c2 = c1 + 1; if (c2 == 3) c2 = 0;
    atomicAdd(&h[c0 * NBINS + binof(a.x)], 1u);
    atomicAdd(&h[c1 * NBINS + binof(a.y)], 1u);
    atomicAdd(&h[c2 * NBINS + binof(a.z)], 1u);
    atomicAdd(&h[c0 * NBINS + binof(a.w)], 1u);
    atomicAdd(&h[3 * NBINS + c0 * NBINS + binof(b.x)], 1u);
    atomicAdd(&h[3 * NBINS + c1 * NBINS + binof(b.y)], 1u);
    atomicAdd(&h[3 * NBINS + c2 * NBINS + binof(b.z)], 1u);
    atomicAdd(&h[3 * NBINS + c0 * NBINS + binof(b.w)], 1u);
  }
  for (int i = (nf4 << 2) + tid; i < n; i += stride) {  // scalar tail (empty for 3*2^24)
    int c = i % 3;
    atomicAdd(&h[c * NBINS + binof(src[i])], 1u);
    atomicAdd(&h[3 * NBINS + c * NBINS + binof(tgt[i])], 1u);
  }
  __syncthreads();
  for (int i = threadIdx.x; i < HIST_WORDS; i += blockDim.x) {
    unsigned v = h[i];
    if (v) atomicAdd(&hist[i], v);
  }
}

__global__ __launch_bounds__(256) void hm_setup(const unsigned* __restrict__ hist,
                                                float* __restrict__ tbl, int npix) {
  __shared__ unsigned scan[NBINS];
  __shared__ float fsrc[NBINS], ftgt[NBINS], pm[NBINS];
  const int t = threadIdx.x;                    // 256 threads, t == bin index
  const float norm = (float)(npix - 1);         // 16777215 -> exact in f32
  float4* tabp = (float4*)(tbl + DXS_F);
  float* edges = tbl + DXS_F + TAB_F;
  for (int ch = 0; ch < 3; ++ch) {
    for (int img = 0; img < 2; ++img) {
      scan[t] = hist[img * 3 * NBINS + ch * NBINS + t];
      __syncthreads();
      for (int off = 1; off < NBINS; off <<= 1) {   // inclusive Hillis-Steele scan
        unsigned v = (t >= off) ? scan[t - off] : 0u;
        __syncthreads();
        scan[t] += v;
        __syncthreads();
      }
      // cdf_min == cdf[0] (hist >= 0 => cdf non-decreasing); diffs <= 2^24 exact in f32
      float f = ((float)(scan[t] - scan[0]) * 2.0f) / norm - 1.0f;
      if (img == 0) fsrc[t] = f; else ftgt[t] = f;
      __syncthreads();
    }
    // pxmap[t] = interpolate(dx=cdf_tgt, dy=grid[512], x=cdf_src[t])
    {
      float x = fsrc[t];
      unsigned j = lb256(ftgt, x);
      int c0 = (int)j - 1; if (c0 < 0) c0 = 0; if (c0 > 255) c0 = 255;
      int c1 = (int)j;     if (c1 > 255) c1 = 255;
      int nr = (fabsf(x - ftgt[c0]) <= fabsf(x - ftgt[c1])) ? c0 : c1;
      unsigned ind1 = lb256(ftgt, ftgt[nr]);        // first occurrence, in [0,255]
      int ind0 = (int)ind1 - 1;
      int i0x = (ind0 < 0) ? ind0 + NBINS : ind0;
      int i0y = (ind0 < 0) ? ind0 + 2 * NBINS : ind0;
      float denom = ftgt[ind1] - ftgt[i0x];
      float tt = (denom == 0.0f) ? 0.0f : ((x - ftgt[i0x]) / denom);
      float gy0 = (float)i0y / 256.0f - 1.0f;       // grid[i] = i/256 - 1 (exact)
      float gy1 = (float)ind1 / 256.0f - 1.0f;
      float mid = gy0 + (gy1 - gy0) * tt;
      pm[t] = (x <= ftgt[0]) ? -1.0f
            : ((x >= ftgt[NBINS - 1]) ? (511.0f / 256.0f - 1.0f) : mid);
    }
    __syncthreads();
    // Per-bin coefficients for the pixel map (dx=cdf_src, dy=pxmap, n==m==256):
    // everything except the initial searchsorted depends only on 'near'.
    {
      unsigned ind1 = lb256(fsrc, fsrc[t]);
      int ind0 = (int)ind1 - 1;
      int i0 = (ind0 < 0) ? ind0 + NBINS : ind0;
      float x0  = fsrc[i0];
      float dn  = fsrc[ind1] - fsrc[i0];
      float dy0 = pm[i0];
      float ddy = pm[ind1] - pm[i0];
      tbl[ch * NBINS + t]  = fsrc[t];               // dx knots for binary search
      tabp[ch * NBINS + t] = make_float4(x0, dn, dy0, ddy);
      if (t == 0) { edges[ch * 2] = pm[0]; edges[ch * 2 + 1] = pm[NBINS - 1]; }
    }
    __syncthreads();                                // before fsrc/ftgt/pm reuse
  }
}

__device__ __forceinline__ float map_one(const float* __restrict__ dx,
                                         const float4* __restrict__ tab,
                                         float x, float dxlo, float dxhi,
                                         float e0, float e1) {
  unsigned j = lb256(dx, x);
  int c0 = (int)j - 1; if (c0 < 0) c0 = 0; if (c0 > 255) c0 = 255;
  int c1 = (int)j;     if (c1 > 255) c1 = 255;
  int nr = (fabsf(x - dx[c0]) <= fabsf(x - dx[c1])) ? c0 : c1;
  float4 tb = tab[nr];                           // {x0, denom, dy0, ddy}
  float tt = (tb.y == 0.0f) ? 0.0f : ((x - tb.x) / tb.y);
  float mid = tb.z + tb.w * tt;
  return (x <= dxlo) ? e0 : ((x >= dxhi) ? e1 : mid);
}

__global__ __launch_bounds__(256) void hm_map(const float* __restrict__ src,
                                              float* __restrict__ out,
                                              const float* __restrict__ tbl,
                                              int n) {
  __shared__ __align__(16) float sh[TBL_F];      // 15392 B: knots | tables | edges
  {
    // Stage all tables into LDS via CDNA5 async global->LDS DMA (ASYNCcnt).
    unsigned ldsbase = (unsigned)(uintptr_t)(&sh[0]);
    unsigned long long gbase = (unsigned long long)(uintptr_t)tbl;
    for (int i = threadIdx.x; i < TBL_F / 4; i += blockDim.x) {
      unsigned voff = (unsigned)i * 16u;
      unsigned laddr = ldsbase + voff;
      asm volatile("global_load_async_to_lds_b128 %0, %1, %2"
                   :: "v"(laddr), "v"(voff), "s"(gbase) : "memory");
    }
    asm volatile("s_wait_asynccnt 0x0" ::: "memory");
  }
  __syncthreads();
  float dxlo[3], dxhi[3], e0[3], e1[3];
#pragma unroll
  for (int c = 0; c < 3; ++c) {
    dxlo[c] = sh[c * NBINS];
    dxhi[c] = sh[c * NBINS + NBINS - 1];
    e0[c]   = sh[DXS_F + TAB_F + c * 2];
    e1[c]   = sh[DXS_F + TAB_F + c * 2 + 1];
  }
  const float4* tab = (const float4*)&sh[DXS_F];
  const int stride = gridDim.x * blockDim.x;
  const int tid = blockIdx.x * blockDim.x + threadIdx.x;
  const int ngroups = n / 12;                    // 12 floats/thread: channels k%3
  const float4* s4 = (const float4*)src;
  float4* o4 = (float4*)out;
  for (int g = tid; g < ngroups; g += stride) {
    float4 v0 = s4[g * 3 + 0], v1 = s4[g * 3 + 1], v2 = s4[g * 3 + 2];
    float r[12] = {v0.x, v0.y, v0.z, v0.w, v1.x, v1.y, v1.z, v1.w,
                   v2.x, v2.y, v2.z, v2.w};
#pragma unroll
    for (int k = 0; k < 12; ++k) {
      const int c = k % 3;
      r[k] = map_one(&sh[c * NBINS], tab + c * NBINS, r[k],
                     dxlo[c], dxhi[c], e0[c], e1[c]);
    }
    o4[g * 3 + 0] = make_float4(r[0], r[1], r[2],  r[3]);
    o4[g * 3 + 1] = make_float4(r[4], r[5], r[6],  r[7]);
    o4[g * 3 + 2] = make_float4(r[8], r[9], r[10], r[11]);
  }
  for (int i = ngroups * 12 + tid; i < n; i += stride) {   // tail (empty here)
    int c = i % 3;
    out[i] = map_one(&sh[c * NBINS], tab + c * NBINS, src[i],
                     dxlo[c], dxhi[c], e0[c], e1[c]);
  }
}

extern "C" void kernel_launch(void* const* d_in, const int* in_sizes, int n_in,
                              void* d_out, int out_size, void* d_ws, size_t ws_size,
                              hipStream_t stream) {
  const float* src = (const float*)d_in[0];
  const float* tgt = (const float*)d_in[1];
  float* out = (float*)d_out;
  unsigned* hist = (unsigned*)d_ws;              // 6144 B
  float* tbl = (float*)d_ws + HIST_WORDS;        // 15392 B, 16B-aligned
  int n = in_sizes[0];                           // 4096*4096*3
  int npix = n / 3;
  hm_zero<<<(HIST_WORDS + 255) / 256, 256, 0, stream>>>(hist);
  hm_hist<<<2048, 256, 0, stream>>>(src, tgt, hist, n);
  hm_setup<<<1, 256, 0, stream>>>(hist, tbl, npix);
  hm_map<<<4096, 256, 0, stream>>>(src, out, tbl, n);
}